// SOFA_15779709846400
// MI455X (gfx1250) — compile-verified
//
#include <hip/hip_runtime.h>
#include <math.h>

typedef __attribute__((ext_vector_type(2))) float v2f;
typedef __attribute__((ext_vector_type(8))) float v8f;

#define N_ELEMS 8192
#define TPB_A   1024
#define ITERS_A (N_ELEMS / TPB_A)   // 8
#define WAVES_B 8                   // waves (i-tiles) per block in attention kernel

// ---------------------------------------------------------------------------
// Phase A (single block): l, angle, mean/std (ddof=1), column gate.
// Packs pk[j] = (l[j], angle[j], c[j], 0) where c[j] = (l<thresh) ? -inf : 0.
// ---------------------------------------------------------------------------
__global__ __launch_bounds__(TPB_A) void sofa_prep(const float* __restrict__ x,
                                                   float4* __restrict__ pk) {
  const int tid  = threadIdx.x;
  const int lane = tid & 31;
  const int wave = tid >> 5;

  __shared__ double s_sum[TPB_A / 32];
  __shared__ double s_sq[TPB_A / 32];
  __shared__ float  s_thresh;

  float  lv[ITERS_A];
  double sum = 0.0, sq = 0.0;

#pragma unroll
  for (int it = 0; it < ITERS_A; ++it) {
    const int j  = tid + it * TPB_A;
    const float x0 = x[2 * j + 0];
    const float x1 = x[2 * j + 1];
    const float l   = sqrtf(fmaf(x0, x0, x1 * x1));
    const float ang = atan2f(x1, x0);
    lv[it] = l;
    pk[j]  = make_float4(l, ang, 0.0f, 0.0f);
    sum += (double)l;
    sq  += (double)l * (double)l;
  }

  // wave32 butterfly reduce, then cross-wave via LDS
#pragma unroll
  for (int off = 16; off > 0; off >>= 1) {
    sum += __shfl_down(sum, off, 32);
    sq  += __shfl_down(sq,  off, 32);
  }
  if (lane == 0) { s_sum[wave] = sum; s_sq[wave] = sq; }
  __syncthreads();

  if (tid == 0) {
    double S = 0.0, Q = 0.0;
    for (int w = 0; w < TPB_A / 32; ++w) { S += s_sum[w]; Q += s_sq[w]; }
    const double mean = S / (double)N_ELEMS;
    const double var  = (Q - (double)N_ELEMS * mean * mean) / (double)(N_ELEMS - 1);
    s_thresh = (float)(mean + 0.1 * sqrt(var));
  }
  __syncthreads();

  const float thresh = s_thresh;
#pragma unroll
  for (int it = 0; it < ITERS_A; ++it) {
    const int j = tid + it * TPB_A;
    pk[j].z = (lv[it] < thresh) ? -__builtin_inff() : 0.0f;
  }
}

// ---------------------------------------------------------------------------
// Phase B: streaming softmax-attention over j, 16-row i-tile per wave.
// WMMA f32 16x16x4 accumulates numerator (col 0) and denominator (col 1).
// ---------------------------------------------------------------------------
__global__ __launch_bounds__(32 * WAVES_B) void sofa_attn(const float4* __restrict__ pk,
                                                          float* __restrict__ out) {
  const int lane    = threadIdx.x & 31;
  const int wave    = threadIdx.x >> 5;
  const int tile    = blockIdx.x * WAVES_B + wave;
  const int rowbase = tile * 16;
  const int m       = lane & 15;          // A: row M ; B: column N
  const int koff    = (lane >> 4) << 1;   // K-pair: lanes 0-15 -> K=0,1 ; 16-31 -> K=2,3

  const float lm = pk[rowbase + m].x;     // this lane's row length (lanes<16: exact row value)

  v8f acc = {0.0f, 0.0f, 0.0f, 0.0f, 0.0f, 0.0f, 0.0f, 0.0f};

#pragma unroll 4
  for (int j0 = 0; j0 < N_ELEMS; j0 += 4) {
    const float4 q0 = pk[j0 + koff + 0];  // (l, angle, c, 0)
    const float4 q1 = pk[j0 + koff + 1];

    const float d0 = lm - q0.x;
    const float d1 = lm - q1.x;
    // p = exp(0.5*d^2 + c); c = -inf zeroes masked columns exactly
    const float a0 = __expf(fmaf(0.5f * d0, d0, q0.z));
    const float a1 = __expf(fmaf(0.5f * d1, d1, q1.z));

    // B[k][n]: col 0 = angle, col 1 = 1.0, rest 0
    const float b0 = (m == 0) ? q0.y : ((m == 1) ? 1.0f : 0.0f);
    const float b1 = (m == 0) ? q1.y : ((m == 1) ? 1.0f : 0.0f);

    v2f A = {a0, a1};
    v2f B = {b0, b1};
    acc = __builtin_amdgcn_wmma_f32_16x16x4_f32(
        /*neg_a=*/false, A, /*neg_b=*/false, B,
        /*c_mod=*/(short)0, acc, /*reuse_a=*/false, /*reuse_b=*/false);
  }

  // D layout: lanes 0-15 -> N=lane, VGPR r -> M=r ; lanes 16-31 -> M=r+8.
  // num = column 0 (lanes 0 & 16), den = column 1 (lanes 1 & 17).
  __shared__ float s_red[WAVES_B][2][16];
  if (lane == 0) {
#pragma unroll
    for (int r = 0; r < 8; ++r) s_red[wave][0][r] = acc[r];
  } else if (lane == 16) {
#pragma unroll
    for (int r = 0; r < 8; ++r) s_red[wave][0][8 + r] = acc[r];
  } else if (lane == 1) {
#pragma unroll
    for (int r = 0; r < 8; ++r) s_red[wave][1][r] = acc[r];
  } else if (lane == 17) {
#pragma unroll
    for (int r = 0; r < 8; ++r) s_red[wave][1][8 + r] = acc[r];
  }
  __syncthreads();

  if (lane < 16) {
    const float num = s_red[wave][0][lane];
    const float den = s_red[wave][1][lane];
    const float oa  = num / den;
    const int row   = rowbase + lane;     // lm == l[row] for lanes < 16
    out[2 * row + 0] = lm * cosf(oa);
    out[2 * row + 1] = lm * sinf(oa);
  }
}

// ---------------------------------------------------------------------------
extern "C" void kernel_launch(void* const* d_in, const int* in_sizes, int n_in,
                              void* d_out, int out_size, void* d_ws, size_t ws_size,
                              hipStream_t stream) {
  const float* x  = (const float*)d_in[0];
  // d_in[1] (w) is unused in the trainable=False reference path.
  float*  out = (float*)d_out;
  float4* pk  = (float4*)d_ws;            // 8192 * 16B = 128 KB scratch

  sofa_prep<<<1, TPB_A, 0, stream>>>(x, pk);

  const int tiles  = N_ELEMS / 16;              // 512 i-tiles
  const int blocks = tiles / WAVES_B;           // 64 blocks x 8 waves
  sofa_attn<<<blocks, 32 * WAVES_B, 0, stream>>>(pk, out);
}